// IntraGNN_81509889343633
// MI455X (gfx1250) — compile-verified
//
#include <hip/hip_runtime.h>

// Problem constants (from reference): M=32 samples, N=1024 nodes, H=64 hidden, C=10 classes
#define MM 32
#define NN 1024
#define HH 64
#define CC 10

// d_out flat layout: feat_s(32*64) | W(32*1024*1024) | sr_s(32*10) | b_s(32*10) | u_s(32)
#define FEAT_OFF 0
#define W_OFF    (MM*HH)                         // 2048
#define SR_OFF   (W_OFF + MM*NN*NN)              // 33556480
#define B_OFF    (SR_OFF + MM*CC)                // 33556800
#define U_OFF    (B_OFF + MM*CC)                 // 33557120

typedef __attribute__((ext_vector_type(2))) float f32x2;
typedef __attribute__((ext_vector_type(8))) float f32x8;

// One wave computes a 16x64 stripe of OUT(NNxHH) = A(NNxNN) @ B(NNxHH) as four
// 16x16 tiles sharing one A fragment per K-step (V_WMMA_F32_16X16X4_F32, exact fp32).
// A layout (16x4 f32): lane = 16*khalf + m ; the 2 VGPRs hold K = khalf*2 + {0,1}
// B layout (4x16 f32): lane = 16*khalf + n ; the 2 VGPRs hold K = khalf*2 + {0,1}
// C/D layout (16x16 f32): lane = 16*half + n ; VGPR v holds row M = 8*half + v
__device__ __forceinline__ void wmma_stripe_nh(const float* __restrict__ A, int lda,
                                               const float* __restrict__ B, int ldb,
                                               int row0, int lane, f32x8 acc[4]) {
  const int nm = lane & 15;
  const int kh = lane >> 4;
  const float* arow = A + (size_t)(row0 + nm) * lda;
  const float* bcol = B + nm;
#pragma unroll 2
  for (int k = 0; k < NN; k += 4) {
    const int ka = k + kh * 2;
    const f32x2 a = *(const f32x2*)(arow + ka);      // one 8B A load feeds 4 WMMAs
#pragma unroll
    for (int ct = 0; ct < 4; ++ct) {
      f32x2 b;
      b.x = bcol[(ka + 0) * ldb + ct * 16];
      b.y = bcol[(ka + 1) * ldb + ct * 16];
      acc[ct] = __builtin_amdgcn_wmma_f32_16x16x4_f32(false, a, false, b, (short)0,
                                                      acc[ct], false, false);
    }
  }
}

// ---------------- Kernel A: XW[m] = X[m] @ Wg  (Wg chosen by device-side view_flag)
__global__ void xw_kernel(const float* __restrict__ X,
                          const float* __restrict__ wg0, const float* __restrict__ wg1,
                          const float* __restrict__ wg2, const int* __restrict__ flag,
                          float* __restrict__ XW) {
  const int vf = *flag;
  const float* Wg = (vf == 1) ? wg1 : ((vf == 2) ? wg2 : wg0);
  const int wave = threadIdx.x >> 5, lane = threadIdx.x & 31;
  const int row0 = (blockIdx.x * 8 + wave) << 4;   // 64 row stripes per sample
  const int m = blockIdx.y;
  const float* A = X + (size_t)m * NN * NN;
  f32x8 acc[4] = {{0.f}, {0.f}, {0.f}, {0.f}};
  wmma_stripe_nh(A, NN, Wg, HH, row0, lane, acc);
  float* out = XW + (size_t)m * NN * HH;
  const int nm = lane & 15, kh = lane >> 4;
#pragma unroll
  for (int ct = 0; ct < 4; ++ct)
#pragma unroll
    for (int v = 0; v < 8; ++v)
      out[(row0 + kh * 8 + v) * HH + ct * 16 + nm] = acc[ct][v];
}

// ---------------- Kernel B: h1 = 0.5*(W0@XW + delta1 (x) XW[0,:]) + XW
// (adjp_1 = 0.5*enh(W0, p=5) + I ; enh only perturbs column 0 -> rank-1 delta term)
__global__ void h1_kernel(const float* __restrict__ Wts, const float* __restrict__ XW,
                          float* __restrict__ H) {
  const int wave = threadIdx.x >> 5, lane = threadIdx.x & 31;
  const int row0 = (blockIdx.x * 8 + wave) << 4;
  const int m = blockIdx.y;
  const float* A  = Wts + (size_t)m * NN * NN;
  const float* xw = XW + (size_t)m * NN * HH;
  f32x8 acc[4] = {{0.f}, {0.f}, {0.f}, {0.f}};
  wmma_stripe_nh(A, NN, xw, HH, row0, lane, acc);
  float* h = H + (size_t)m * NN * HH;
  const int nm = lane & 15, kh = lane >> 4;
#pragma unroll
  for (int ct = 0; ct < 4; ++ct)
#pragma unroll
    for (int v = 0; v < 8; ++v) {
      const int i = row0 + kh * 8 + v;
      const int j = ct * 16 + nm;
      float val = 0.5f * acc[ct][v] + xw[i * HH + j];  // +XW is the identity term
      if (i < 5) {                                     // p = 5*t with t=1
        const float w00 = A[(size_t)i * NN];           // W0[i][0]
        const float d = (w00 > 0.f) ? (1.f - w00) : 0.f;
        val += 0.5f * d * xw[j];                       // delta * XW[0][j]
      }
      h[i * HH + j] = val;
    }
}

__device__ __forceinline__ float softplusf(float x) {
  return (x > 20.f) ? x : log1pf(expf(x));
}

// ---------------- Kernel C: persistent per-sample loop (1 block per sample).
// State: h (global, L2-resident), col0 vector c (LDS). Each iteration:
//   feat = mean(leaky(h)) ; sr = softplus(feat@fc_w+fc_b) ; u = C/S ; break test
//   store outputs ; c <- 0.5*enh(c,5t)+e0 ; h <- 0.5*(h + delta (x) XW0) + XW
__global__ void loop_kernel(const float* __restrict__ Wts,
                            const float* __restrict__ fc_w, const float* __restrict__ fc_b,
                            const float* __restrict__ XW, float* __restrict__ H,
                            float* __restrict__ c_out, int* __restrict__ T_out,
                            float* __restrict__ out) {
  __shared__ float sh_c[NN];
  __shared__ float sh_part[256];
  __shared__ float sh_feat[HH];
  __shared__ float sh_sr[CC];
  __shared__ float sh_xw0[HH];
  __shared__ float sh_Su[2];
  __shared__ int sh_brk;
  const int m = blockIdx.x, tid = threadIdx.x;
  const float* xw = XW + (size_t)m * NN * HH;
  float* h = H + (size_t)m * NN * HH;

  for (int i = tid; i < NN; i += 256)
    sh_c[i] = Wts[(size_t)m * NN * NN + (size_t)i * NN];   // W0[:,0]
  if (tid < HH) sh_xw0[tid] = xw[tid];                     // XW[0,:]
  __syncthreads();

  float u0 = 999.f;   // only thread 0's copy is authoritative
  int Tst = 0;
  for (int t = 1; t <= 8; ++t) {
    // --- feat = mean over rows of leakyrelu(h); coalesced: col = tid%64 fixed
    float p = 0.f;
    for (int k = 0; k < (NN * HH) / 256; ++k) {
      const float v = h[k * 256 + tid];
      p += (v >= 0.f) ? v : 0.2f * v;
    }
    sh_part[tid] = p;
    __syncthreads();
    if (tid < HH)
      sh_feat[tid] = (sh_part[tid] + sh_part[tid + 64] + sh_part[tid + 128] +
                      sh_part[tid + 192]) * (1.f / NN);
    __syncthreads();
    if (tid < CC) {
      float a = fc_b[tid];
      for (int j = 0; j < HH; ++j) a += sh_feat[j] * fc_w[j * CC + tid];
      sh_sr[tid] = softplusf(a);
    }
    __syncthreads();
    if (tid == 0) {
      float S = (float)CC;
      for (int c = 0; c < CC; ++c) S += sh_sr[c];
      const float u = (float)CC / S;
      const int brk = (u >= u0) || (t == 8);     // 5*t>=40 <=> t==8
      sh_Su[0] = S; sh_Su[1] = u;
      sh_brk = brk;
      if (!brk) { u0 = u; Tst = t; }
    }
    __syncthreads();
    if (sh_brk) break;

    // --- store outputs (last store wins, exactly like the masked reference)
    const float S = sh_Su[0];
    if (tid < HH) out[FEAT_OFF + m * HH + tid] = sh_feat[tid];
    if (tid < CC) {
      out[SR_OFF + m * CC + tid] = sh_sr[tid];
      out[B_OFF + m * CC + tid] = sh_sr[tid] / S;
    }
    if (tid == 0) out[U_OFF + m] = sh_Su[1];

    // --- c <- col0 of adjp_t = 0.5*enh(c, 5t) + e0
    for (int i = tid; i < NN; i += 256) {
      float cv = sh_c[i];
      if (i < 5 * t && cv > 0.f) cv = 1.f;
      cv = 0.5f * cv + (i == 0 ? 1.f : 0.f);
      sh_c[i] = cv;
    }
    __syncthreads();

    // --- h <- h_{t+1} = 0.5*(h + delta_{t+1} (x) XW0) + XW  (O(N*H), no GEMM)
    const int thr = 5 * (t + 1);
    for (int k = 0; k < (NN * HH) / 256; ++k) {
      const int idx = k * 256 + tid;
      const int i = idx >> 6;
      const float cv = sh_c[i];
      const float d = (i < thr && cv > 0.f) ? (1.f - cv) : 0.f;
      h[idx] = 0.5f * (h[idx] + d * sh_xw0[idx & 63]) + xw[idx];
    }
    __syncthreads();
  }

  if (tid == 0) T_out[m] = Tst;
  for (int i = tid; i < NN; i += 256) c_out[m * NN + i] = sh_c[i];
}

// ---------------- Kernel D: reconstruct final W (pure bandwidth, float4).
// W_T[i,j] = 0.5^T * W0[i,j]            (j!=0, i!=j)
//          + 2*(1-0.5^T) on diagonal    (i==j, i>=1)
// W_T[i,0] = tracked col0 recursion c_T[i]   ((0,0) included in c)
__global__ void wout_kernel(const float* __restrict__ Wts, const float* __restrict__ c_in,
                            const int* __restrict__ T_in, float* __restrict__ outW) {
  const int gid = blockIdx.x * 256 + threadIdx.x;      // one float4 each
  const int m = gid >> 18;                             // 262144 float4 / sample
  const int rem = gid & ((1 << 18) - 1);
  const int i = rem >> 8;                              // 256 float4 / row
  const int j = (rem & 255) << 2;
  const size_t off = (size_t)m * NN * NN + (size_t)i * NN + j;
  __builtin_prefetch(Wts + off + 2048, 0, 1);          // global_prefetch_b8
  const float4 w = *(const float4*)(Wts + off);
  const int T = T_in[m];
  const float s = exp2f(-(float)T);
  const float dg = 2.f * (1.f - s);
  float r0 = w.x * s, r1 = w.y * s, r2 = w.z * s, r3 = w.w * s;
  if (j == 0) r0 = c_in[m * NN + i];                   // column 0 override
  const int dv = i - j;
  if (dv >= 0 && dv < 4 && i != 0) {                   // diagonal (not (0,0))
    if (dv == 0) r0 += dg; else if (dv == 1) r1 += dg;
    else if (dv == 2) r2 += dg; else r3 += dg;
  }
  *(float4*)(outW + off) = make_float4(r0, r1, r2, r3);
}

extern "C" void kernel_launch(void* const* d_in, const int* in_sizes, int n_in,
                              void* d_out, int out_size, void* d_ws, size_t ws_size,
                              hipStream_t stream) {
  const float* features = (const float*)d_in[0];
  const float* weights  = (const float*)d_in[1];
  const float* wg0      = (const float*)d_in[2];
  const float* wg1      = (const float*)d_in[3];
  const float* wg2      = (const float*)d_in[4];
  const float* fc_w     = (const float*)d_in[5];
  const float* fc_b     = (const float*)d_in[6];
  const int*   flag     = (const int*)d_in[7];
  float* out = (float*)d_out;

  // workspace: XW (8MB) | h (8MB) | c_final (128KB) | T (128B)
  float* ws_xw = (float*)d_ws;
  float* ws_h  = ws_xw + (size_t)MM * NN * HH;
  float* ws_c  = ws_h + (size_t)MM * NN * HH;
  int*   ws_T  = (int*)(ws_c + (size_t)MM * NN);

  dim3 gGemm(8, MM);   // 8 waves/block, each wave one 16x64 stripe -> 64 stripes/sample
  xw_kernel<<<gGemm, 256, 0, stream>>>(features, wg0, wg1, wg2, flag, ws_xw);
  h1_kernel<<<gGemm, 256, 0, stream>>>(weights, ws_xw, ws_h);
  loop_kernel<<<MM, 256, 0, stream>>>(weights, fc_w, fc_b, ws_xw, ws_h, ws_c, ws_T, out);
  wout_kernel<<<(MM * NN * NN / 4) / 256, 256, 0, stream>>>(weights, ws_c, ws_T,
                                                            out + W_OFF);
}